// GNNModel_74766790689426
// MI455X (gfx1250) — compile-verified
//
#include <hip/hip_runtime.h>

// Hetero GraphSAGE for MI455X (gfx1250, wave32).
// Bottleneck analysis: 3x 300k-edge gather + scatter-add (L2-resident, 192MB L2
// holds both 51MB feature tables) >> 20 GFLOP of GEMM. So: fp32 end-to-end,
// GEMMs on the matrix cores via V_WMMA_F32_16X16X4_F32 (exact fp32 math).

typedef __attribute__((ext_vector_type(2))) float v2f;
typedef __attribute__((ext_vector_type(8))) float v8f;

#define FEAT 128

// ---------------------------------------------------------------- zero scratch
__global__ __launch_bounds__(256) void zero_kernel(float* __restrict__ p, long n) {
  long n4 = n >> 2;
  long i = (long)blockIdx.x * blockDim.x + threadIdx.x;
  long stride = (long)gridDim.x * blockDim.x;
  float4 z = make_float4(0.f, 0.f, 0.f, 0.f);
  for (long j = i; j < n4; j += stride) ((float4*)p)[j] = z;
  if (i == 0) {
    for (long j = (n4 << 2); j < n; ++j) p[j] = 0.f;
  }
}

// ------------------------------------------------- scatter-add (segment sums)
// One wave32 per edge: each lane moves 4 floats (float4 load -> 4 f32 atomics).
__global__ __launch_bounds__(256) void scatter_kernel(
    const float* __restrict__ xsrc, const int* __restrict__ ei, // [2,E]
    float* __restrict__ sums, float* __restrict__ cnt, int E) {
  int gid  = blockIdx.x * blockDim.x + threadIdx.x;
  int edge = gid >> 5;
  int lane = gid & 31;
  if (edge >= E) return;
  int src = ei[edge];
  int dst = ei[E + edge];
  const float4 v = *(const float4*)(xsrc + (size_t)src * FEAT + lane * 4);
  float* s = sums + (size_t)dst * FEAT + lane * 4;
  atomicAdd(s + 0, v.x);
  atomicAdd(s + 1, v.y);
  atomicAdd(s + 2, v.z);
  atomicAdd(s + 3, v.w);
  if (lane == 0) atomicAdd(cnt + dst, 1.0f);
}

// ------------------------------------------------------ fused SAGE GEMM (WMMA)
// out[16 rows x 128 cols] per block. 8 waves; wave w owns N-tile [16w,16w+16).
// acc = (sums/max(cnt,1)) @ Wl^T + x_dst @ Wr^T  (+ bl at store).
// fp32 WMMA fragment layout (ISA 7.12.2):
//   A 16x4:  lane L -> M = L%16, VGPR{0,1} hold K = k0 + 2*(L/16) + {0,1}
//   B 4x16:  lane L -> N = L%16, VGPR{0,1} hold K = k0 + 2*(L/16) + {0,1}
//   C 16x16: VGPR r -> M = r + 8*(L/16), N = L%16
__global__ __launch_bounds__(256) void sage_gemm_kernel(
    const float* __restrict__ sums, const float* __restrict__ cnt,
    const float* __restrict__ xdst,
    const float* __restrict__ Wl,  // [H=128, D=128] row-major
    const float* __restrict__ bl,  // [128]
    const float* __restrict__ Wr,  // [H=128, D=128] row-major
    float* __restrict__ out,       // [n, 128]
    int n, int accum) {
  const int lane = threadIdx.x & 31;
  const int wave = threadIdx.x >> 5;  // 0..7 -> N tile
  const int lm   = lane & 15;
  const int half = lane >> 4;
  const int m0   = blockIdx.x * 16;

  int rowA = m0 + lm;
  if (rowA >= n) rowA = n - 1;  // keep EXEC all-ones (WMMA requirement)
  const int ncol = wave * 16 + lm;

  const float c   = cnt[rowA];
  const float inv = 1.0f / fmaxf(c, 1.0f);

  const float* arow_mean = sums + (size_t)rowA * FEAT + 2 * half;
  const float* arow_x    = xdst + (size_t)rowA * FEAT + 2 * half;
  const float* brow_l    = Wl   + (size_t)ncol * FEAT + 2 * half;
  const float* brow_r    = Wr   + (size_t)ncol * FEAT + 2 * half;

  v8f acc = {};
#pragma unroll
  for (int k0 = 0; k0 < FEAT; k0 += 4) {
    v2f a = *(const v2f*)(arow_mean + k0);
    a.x *= inv;
    a.y *= inv;
    v2f b = *(const v2f*)(brow_l + k0);
    acc = __builtin_amdgcn_wmma_f32_16x16x4_f32(false, a, false, b,
                                                (short)0, acc, false, false);
  }
#pragma unroll
  for (int k0 = 0; k0 < FEAT; k0 += 4) {
    v2f a = *(const v2f*)(arow_x + k0);
    v2f b = *(const v2f*)(brow_r + k0);
    acc = __builtin_amdgcn_wmma_f32_16x16x4_f32(false, a, false, b,
                                                (short)0, acc, false, false);
  }

  const float bias = bl[ncol];
#pragma unroll
  for (int r = 0; r < 8; ++r) {
    int row = m0 + r + 8 * half;
    if (row < n) {
      float v = acc[r] + bias;
      size_t idx = (size_t)row * FEAT + ncol;
      out[idx] = accum ? (out[idx] + v) : v;
    }
  }
}

// ---------------------------------------------------------------------- launch
static inline void run_pass(const float* xsrc, const float* xdst, const int* ei,
                            int E, int n_dst, const float* Wl, const float* bl,
                            const float* Wr, float* out, int accum,
                            float* ws, hipStream_t stream) {
  float* sums = ws;
  float* cnt  = ws + (size_t)n_dst * FEAT;
  long nzero = (long)n_dst * (FEAT + 1);
  zero_kernel<<<1024, 256, 0, stream>>>(sums, nzero);
  scatter_kernel<<<(E + 7) / 8, 256, 0, stream>>>(xsrc, ei, sums, cnt, E);
  sage_gemm_kernel<<<(n_dst + 15) / 16, 256, 0, stream>>>(
      sums, cnt, xdst, Wl, bl, Wr, out, n_dst, accum);
}

extern "C" void kernel_launch(void* const* d_in, const int* in_sizes, int n_in,
                              void* d_out, int out_size, void* d_ws, size_t ws_size,
                              hipStream_t stream) {
  const float* x_svc      = (const float*)d_in[0];
  const float* x_ep       = (const float*)d_in[1];
  const int*   ei_calls   = (const int*)d_in[2];
  const int*   ei_has     = (const int*)d_in[3];
  const int*   ei_belongs = (const int*)d_in[4];
  const float* Wl_calls   = (const float*)d_in[5];
  const float* bl_calls   = (const float*)d_in[6];
  const float* Wr_calls   = (const float*)d_in[7];
  const float* Wl_has     = (const float*)d_in[8];
  const float* bl_has     = (const float*)d_in[9];
  const float* Wr_has     = (const float*)d_in[10];
  const float* Wl_belongs = (const float*)d_in[11];
  const float* bl_belongs = (const float*)d_in[12];
  const float* Wr_belongs = (const float*)d_in[13];

  const int n_svc = in_sizes[0] / FEAT;
  const int n_ep  = in_sizes[1] / FEAT;
  const int E     = in_sizes[2] / 2;

  float* out_svc = (float*)d_out;
  float* out_ep  = out_svc + (size_t)n_svc * FEAT;
  float* ws      = (float*)d_ws;

  // (service -calls-> service): writes out_service
  run_pass(x_svc, x_svc, ei_calls, E, n_svc,
           Wl_calls, bl_calls, Wr_calls, out_svc, /*accum=*/0, ws, stream);
  // (endpoint -belongs-> service): accumulates into out_service
  run_pass(x_ep, x_svc, ei_belongs, E, n_svc,
           Wl_belongs, bl_belongs, Wr_belongs, out_svc, /*accum=*/1, ws, stream);
  // (service -has-> endpoint): writes out_endpoint
  run_pass(x_svc, x_ep, ei_has, E, n_ep,
           Wl_has, bl_has, Wr_has, out_ep, /*accum=*/0, ws, stream);
}